// TopOptGNN_14697378087057
// MI455X (gfx1250) — compile-verified
//
#include <hip/hip_runtime.h>
#include <math.h>

typedef float v2f __attribute__((ext_vector_type(2)));
typedef float v8f __attribute__((ext_vector_type(8)));

#define HG 512
#define WG 512
#define NN (HG * WG)
#define F  64
#define PR 160   // padded pair-row stride (floats) in LDS: 128 data + 32 pad

// deg(i,j) = 1 (self loop) + in-bounds 4-neighbors; dinv = deg^-1/2
__device__ __forceinline__ float dinv_of(int node) {
    int i = node >> 9;          // /512
    int j = node & (WG - 1);    // %512
    float deg = 1.0f + (i > 0) + (i < HG - 1) + (j > 0) + (j < WG - 1);
    return rsqrtf(deg);
}

// ---------------------------------------------------------------------------
// Kernel 1: g1[v][:] = (x[v] @ W1) * dinv[v]      (x: [N,2], W1: [2,64])
// ---------------------------------------------------------------------------
__global__ __launch_bounds__(256) void k_g1(const float* __restrict__ x,
                                            const float* __restrict__ W1,
                                            float* __restrict__ g1) {
    int idx = blockIdx.x * blockDim.x + threadIdx.x;
    if (idx >= NN * (F / 4)) return;
    int node = idx >> 4;
    int c = (idx & 15) << 2;
    float x0 = x[node * 2 + 0];
    float x1 = x[node * 2 + 1];
    float4 w0 = *(const float4*)(W1 + c);        // W1 row 0
    float4 w1 = *(const float4*)(W1 + F + c);    // W1 row 1
    float di = dinv_of(node);
    float4 g;
    g.x = (x0 * w0.x + x1 * w1.x) * di;
    g.y = (x0 * w0.y + x1 * w1.y) * di;
    g.z = (x0 * w0.z + x1 * w1.z) * di;
    g.w = (x0 * w0.w + x1 * w1.w) * di;
    *(float4*)(g1 + (size_t)node * F + c) = g;
}

// ---------------------------------------------------------------------------
// Kernel 2: h[v] = relu(dinv[v] * (5-point stencil sum of g) + b)
// ---------------------------------------------------------------------------
__device__ __forceinline__ void add4(float4& a, const float4 b) {
    a.x += b.x; a.y += b.y; a.z += b.z; a.w += b.w;
}

__global__ __launch_bounds__(256) void k_stencil_relu(const float* __restrict__ g,
                                                      const float* __restrict__ b,
                                                      float* __restrict__ h) {
    int idx = blockIdx.x * blockDim.x + threadIdx.x;
    if (idx >= NN * (F / 4)) return;
    int node = idx >> 4;
    int c = (idx & 15) << 2;
    int i = node >> 9;
    int j = node & (WG - 1);
    const float* p = g + (size_t)node * F + c;
    float4 s = *(const float4*)p;
    if (i > 0)      add4(s, *(const float4*)(p - WG * F));
    if (i < HG - 1) add4(s, *(const float4*)(p + WG * F));
    if (j > 0)      add4(s, *(const float4*)(p - F));
    if (j < WG - 1) add4(s, *(const float4*)(p + F));
    float di = dinv_of(node);
    float4 bb = *(const float4*)(b + c);
    float4 o;
    o.x = fmaxf(di * s.x + bb.x, 0.0f);
    o.y = fmaxf(di * s.y + bb.y, 0.0f);
    o.z = fmaxf(di * s.z + bb.z, 0.0f);
    o.w = fmaxf(di * s.w + bb.w, 0.0f);
    *(float4*)(h + (size_t)node * F + c) = o;
}

// ---------------------------------------------------------------------------
// Kernel 3: g2 = (h1 @ W2) * dinv  via V_WMMA_F32_16X16X4_F32.
// Block = 256 threads = 8 waves; each wave computes a 16x64 output tile
// (four 16x16 accumulator tiles), K=64 in 16 steps of K=4.
//
// A layout (16x4 f32): lanes 0-15 hold K={0,1} in v0/v1 for row M=lane%16;
// lanes 16-31 hold K={2,3}.  B mirrors (lane%16 = N column, half selects the
// K pair), so each B fragment is the pair {W2[br][c], W2[br+1][c]}.
//
// W2 is staged in LDS pair-interleaved: pairs contiguous -> one ds_load_b64
// per fragment, no repacking movs. Pair-row stride = 160 floats so the two
// half-waves (pair rows 2k vs 2k+1, +640B = +32 banks) hit disjoint bank sets.
// ---------------------------------------------------------------------------
__global__ __launch_bounds__(256) void k_gemm_wmma(const float* __restrict__ h1,
                                                   const float* __restrict__ W2,
                                                   float* __restrict__ g2) {
    __shared__ float W2p[32 * PR];  // 20 KB, pair-interleaved + padded
    for (int t = threadIdx.x; t < F * F; t += 256) {
        int r = t >> 6;     // source row 0..63
        int c = t & 63;     // source col
        W2p[(r >> 1) * PR + c * 2 + (r & 1)] = W2[t];
    }
    __syncthreads();

    int wave = threadIdx.x >> 5;
    int lane = threadIdx.x & 31;
    int half = lane >> 4;   // selects K pair {0,1} vs {2,3} within each K=4 step
    int lm   = lane & 15;

    int row_base = (blockIdx.x * 8 + wave) * 16;
    const float* arow = h1 + (size_t)(row_base + lm) * F + 2 * half;

    v8f c0 = {}, c1 = {}, c2 = {}, c3 = {};

#pragma unroll
    for (int k = 0; k < 16; ++k) {
        v2f a = *(const v2f*)(arow + 4 * k);            // h1[row][4k+2*half+{0,1}]
        const float* bp = &W2p[(2 * k + half) * PR + 2 * lm];
        v2f b0 = *(const v2f*)(bp);                      // N-tile 0
        v2f b1 = *(const v2f*)(bp + 32);                 // N-tile 1 (+16 cols)
        v2f b2 = *(const v2f*)(bp + 64);                 // N-tile 2
        v2f b3 = *(const v2f*)(bp + 96);                 // N-tile 3
        c0 = __builtin_amdgcn_wmma_f32_16x16x4_f32(false, a, false, b0, (short)0, c0, false, false);
        c1 = __builtin_amdgcn_wmma_f32_16x16x4_f32(false, a, false, b1, (short)0, c1, false, false);
        c2 = __builtin_amdgcn_wmma_f32_16x16x4_f32(false, a, false, b2, (short)0, c2, false, false);
        c3 = __builtin_amdgcn_wmma_f32_16x16x4_f32(false, a, false, b3, (short)0, c3, false, false);
    }

    // C/D layout: VGPR r, lanes 0-15 -> M = r, lanes 16-31 -> M = r + 8; N = lane%16.
#pragma unroll
    for (int r = 0; r < 8; ++r) {
        int row = row_base + r + 8 * half;
        float s = dinv_of(row);
        float* orow = g2 + (size_t)row * F + lm;
        orow[0]  = c0[r] * s;
        orow[16] = c1[r] * s;
        orow[32] = c2[r] * s;
        orow[48] = c3[r] * s;
    }
}

// ---------------------------------------------------------------------------
// Kernel 4: out[v] = sigmoid( relu(dinv*stencil(g2)+b2) . Wfc + bfc )
// One wave per node; lane l handles features {2l, 2l+1}; shfl_xor reduction.
// ---------------------------------------------------------------------------
__global__ __launch_bounds__(256) void k_final(const float* __restrict__ g2,
                                               const float* __restrict__ b2,
                                               const float* __restrict__ Wfc,
                                               const float* __restrict__ bfc,
                                               float* __restrict__ out) {
    int node = (blockIdx.x * blockDim.x + threadIdx.x) >> 5;
    int lane = threadIdx.x & 31;
    if (node >= NN) return;
    int i = node >> 9;
    int j = node & (WG - 1);
    int c = lane * 2;
    const float* p = g2 + (size_t)node * F + c;
    float sx = p[0], sy = p[1];
    if (i > 0)      { sx += p[-WG * F]; sy += p[-WG * F + 1]; }
    if (i < HG - 1) { sx += p[ WG * F]; sy += p[ WG * F + 1]; }
    if (j > 0)      { sx += p[-F];      sy += p[-F + 1]; }
    if (j < WG - 1) { sx += p[ F];      sy += p[ F + 1]; }
    float di = dinv_of(node);
    float h0 = fmaxf(di * sx + b2[c], 0.0f);
    float h1 = fmaxf(di * sy + b2[c + 1], 0.0f);
    float partial = h0 * Wfc[c] + h1 * Wfc[c + 1];
#pragma unroll
    for (int m = 16; m >= 1; m >>= 1) partial += __shfl_xor(partial, m, 32);
    if (lane == 0) {
        float z = partial + bfc[0];
        out[node] = 1.0f / (1.0f + expf(-z));
    }
}

// ---------------------------------------------------------------------------
extern "C" void kernel_launch(void* const* d_in, const int* in_sizes, int n_in,
                              void* d_out, int out_size, void* d_ws, size_t ws_size,
                              hipStream_t stream) {
    (void)in_sizes; (void)n_in; (void)out_size; (void)ws_size;
    const float* x   = (const float*)d_in[0];
    // d_in[1] = edge_index (unused: grid structure is known statically)
    const float* W1  = (const float*)d_in[2];
    const float* b1  = (const float*)d_in[3];
    const float* W2  = (const float*)d_in[4];
    const float* b2  = (const float*)d_in[5];
    const float* Wfc = (const float*)d_in[6];
    const float* bfc = (const float*)d_in[7];
    float* out = (float*)d_out;

    float* gbuf = (float*)d_ws;                  // N*64 floats (g1, then g2)
    float* hbuf = gbuf + (size_t)NN * F;         // N*64 floats (h1)

    const int elemBlocks = (NN * (F / 4) + 255) / 256;   // 16384

    // 1) g1 = (x@W1) * dinv
    k_g1<<<elemBlocks, 256, 0, stream>>>(x, W1, gbuf);
    // 2) h1 = relu(dinv * stencil(g1) + b1)
    k_stencil_relu<<<elemBlocks, 256, 0, stream>>>(gbuf, b1, hbuf);
    // 3) g2 = (h1 @ W2) * dinv      [WMMA f32 16x16x4]
    k_gemm_wmma<<<NN / (16 * 8), 256, 0, stream>>>(hbuf, W2, gbuf);
    // 4) out = sigmoid(relu(dinv*stencil(g2)+b2) . Wfc + bfc)
    k_final<<<(NN * 32) / 256, 256, 0, stream>>>(gbuf, b2, Wfc, bfc, out);
}